// Sampler_27470610825696
// MI455X (gfx1250) — compile-verified
//
#include <hip/hip_runtime.h>
#include <hip/hip_bf16.h>

typedef float v4f __attribute__((ext_vector_type(4)));

#define BATCH         128
#define VOCAB         128000
#define ROWPAIRS      64            // rows b and b+64 share one threefry eval
#define HALF_N        8192000u      // 64 * 128000 = N/2 for the threefry split
#define THREADS       256
#define CHUNKS        25            // 32000 float4 per row / (256 thr * 5 f4)
#define V4_PER_BLOCK  1280
#define V4_PER_THREAD 5

__device__ __forceinline__ unsigned rotl32(unsigned x, unsigned r) {
    return (x << r) | (x >> (32u - r));
}

// Exact JAX threefry2x32 with key = jax.random.key(42) -> (k0,k1) = (0,42).
__device__ __forceinline__ void threefry2x32_key42(unsigned x0, unsigned x1,
                                                   unsigned &o0, unsigned &o1) {
    const unsigned ks0 = 0u;
    const unsigned ks1 = 42u;
    const unsigned ks2 = 0x1BD11BDAu ^ 0u ^ 42u;
    x0 += ks0; x1 += ks1;
#define TF_ROUND(r) { x0 += x1; x1 = rotl32(x1, (r)); x1 ^= x0; }
    TF_ROUND(13) TF_ROUND(15) TF_ROUND(26) TF_ROUND(6)
    x0 += ks1; x1 += ks2 + 1u;
    TF_ROUND(17) TF_ROUND(29) TF_ROUND(16) TF_ROUND(24)
    x0 += ks2; x1 += ks0 + 2u;
    TF_ROUND(13) TF_ROUND(15) TF_ROUND(26) TF_ROUND(6)
    x0 += ks0; x1 += ks1 + 3u;
    TF_ROUND(17) TF_ROUND(29) TF_ROUND(16) TF_ROUND(24)
    x0 += ks1; x1 += ks2 + 4u;
    TF_ROUND(13) TF_ROUND(15) TF_ROUND(26) TF_ROUND(6)
    x0 += ks2; x1 += ks0 + 5u;
#undef TF_ROUND
    o0 = x0; o1 = x1;
}

// Monotone packing: float order -> unsigned order; ~idx so that among equal
// scores the LARGEST packed value has the SMALLEST index (jnp.argmax tiebreak).
__device__ __forceinline__ unsigned long long packKey(float s, int idx) {
    unsigned b = __float_as_uint(s);
    b = (b & 0x80000000u) ? ~b : (b | 0x80000000u);
    return ((unsigned long long)b << 32) | (unsigned)(~idx);
}

// gfx1250 async copy: global -> LDS, GVS addressing (SGPR base + per-lane
// 32-bit byte offset). LDS destination address comes from a per-lane VGPR;
// low 32 bits of a generic pointer to __shared__ are the group-segment offset.
__device__ __forceinline__ void async_stage_b128(const float* __restrict__ base,
                                                 unsigned byteOff,
                                                 v4f* ldsDst) {
    unsigned ldsOff = (unsigned)(uintptr_t)ldsDst;
    asm volatile("global_load_async_to_lds_b128 %0, %1, %2"
                 :
                 : "v"(ldsOff), "v"(byteOff), "s"(base)
                 : "memory");
}

#define WAIT_ASYNCCNT(n) asm volatile("s_wait_asynccnt %0" :: "i"(n) : "memory")
#define FENCE_DSCNT()    asm volatile("s_wait_dscnt 0x0" ::: "memory")

__global__ __launch_bounds__(128)
void sampler_init_ws(unsigned long long* __restrict__ ws) {
    ws[threadIdx.x] = 0ull;
}

__global__ __launch_bounds__(THREADS)
void sampler_gumbel_argmax(const float* __restrict__ logits,
                           const float* __restrict__ temps,
                           unsigned long long* __restrict__ ws) {
    // [buffer][row][tid] : 2 * 2 * 256 * 16B = 16 KB
    __shared__ v4f stage[2][2][THREADS];

    const int chunk = blockIdx.x;        // 0..CHUNKS-1
    const int b     = blockIdx.y;        // 0..63, handles rows b and b+64
    const int tid   = threadIdx.x;

    const float invT0 = 1.0f / temps[b];
    const float invT1 = 1.0f / temps[b + ROWPAIRS];

    const float* __restrict__ row0 = logits + (size_t)b * VOCAB;
    const float* __restrict__ row1 = logits + (size_t)(b + ROWPAIRS) * VOCAB;

    const unsigned baseFlat = (unsigned)b * (unsigned)VOCAB;  // flat idx of (b,0)
    const int v4base = chunk * V4_PER_BLOCK + tid;

    // Prologue: stage tile 0 while we set up.
    async_stage_b128(row0, (unsigned)(v4base * 16), &stage[0][0][tid]);
    async_stage_b128(row1, (unsigned)(v4base * 16), &stage[0][1][tid]);

    float best0 = -INFINITY, best1 = -INFINITY;
    int   idx0  = 0,         idx1  = 0;

#pragma unroll
    for (int j = 0; j < V4_PER_THREAD; ++j) {
        if (j + 1 < V4_PER_THREAD) {
            // WAR fence: previous ds_loads from this buffer must retire
            // before the async engine may overwrite it.
            FENCE_DSCNT();
            const unsigned off = (unsigned)((v4base + (j + 1) * THREADS) * 16);
            async_stage_b128(row0, off, &stage[(j + 1) & 1][0][tid]);
            async_stage_b128(row1, off, &stage[(j + 1) & 1][1][tid]);
            WAIT_ASYNCCNT(2);   // allow next tile in flight; tile j is complete
        } else {
            WAIT_ASYNCCNT(0);   // last tile
        }

        const v4f l0 = stage[j & 1][0][tid];   // ds_load_b128, lane-private slice
        const v4f l1 = stage[j & 1][1][tid];
        const int v = (v4base + j * THREADS) * 4;
#pragma unroll
        for (int c = 0; c < 4; ++c) {
            const unsigned cnt = baseFlat + (unsigned)(v + c);
            unsigned r0, r1;
            // out0 -> bits for (b, v+c); out1 -> bits for (b+64, v+c)
            threefry2x32_key42(cnt, cnt + HALF_N, r0, r1);
            const float u0 = __uint_as_float((r0 >> 9) | 0x3F800000u) - 1.0f;
            const float u1 = __uint_as_float((r1 >> 9) | 0x3F800000u) - 1.0f;
            const float e0 = -log1pf(-u0);     // Exp(1) noise; log1p precision is
            const float e1 = -log1pf(-u1);     // critical for small u (the winners)
            const float s0 = l0[c] * invT0 - __logf(e0);  // hw v_log_f32 is enough here
            const float s1 = l1[c] * invT1 - __logf(e1);
            if (s0 > best0) { best0 = s0; idx0 = v + c; }
            if (s1 > best1) { best1 = s1; idx1 = v + c; }
        }
    }

    // wave32 reduction on sortable packed keys
    unsigned long long p0 = packKey(best0, idx0);
    unsigned long long p1 = packKey(best1, idx1);
#pragma unroll
    for (int off = 16; off > 0; off >>= 1) {
        unsigned long long q0 = __shfl_xor(p0, off, 32);
        unsigned long long q1 = __shfl_xor(p1, off, 32);
        p0 = (q0 > p0) ? q0 : p0;
        p1 = (q1 > p1) ? q1 : p1;
    }
    if ((tid & 31) == 0) {
        atomicMax(&ws[b],            p0);
        atomicMax(&ws[b + ROWPAIRS], p1);
    }
}

__global__ __launch_bounds__(128)
void sampler_finalize(const unsigned long long* __restrict__ ws,
                      float* __restrict__ out) {
    const int i = threadIdx.x;
    const unsigned low = (unsigned)(ws[i] & 0xFFFFFFFFull);
    out[i] = (float)(int)(~low);   // undo ~idx packing
}

extern "C" void kernel_launch(void* const* d_in, const int* in_sizes, int n_in,
                              void* d_out, int out_size, void* d_ws, size_t ws_size,
                              hipStream_t stream) {
    const float* logits = (const float*)d_in[0];   // [128, 128000] f32
    const float* temps  = (const float*)d_in[1];   // [128] f32
    float* out = (float*)d_out;                    // 128 sampled token ids
    unsigned long long* ws = (unsigned long long*)d_ws;  // 128 packed argmax cells

    sampler_init_ws<<<1, BATCH, 0, stream>>>(ws);
    dim3 grid(CHUNKS, ROWPAIRS);
    sampler_gumbel_argmax<<<grid, THREADS, 0, stream>>>(logits, temps, ws);
    sampler_finalize<<<1, BATCH, 0, stream>>>(ws, out);
}